// NeuralGraphPool_52072183497147
// MI455X (gfx1250) — compile-verified
//
#include <hip/hip_runtime.h>

// Problem constants from the reference: B=128, A=128, D=8, F=256 (BF unused).
#define NB 128
#define NA 128
#define ND 8
#define NF 256
#define NF4 (NF / 4)          // 64 float4 per feature row
#define ROWS_PER_BLOCK 4

static __device__ __forceinline__ float4 f4max(float4 a, float4 b) {
    float4 r;
    r.x = fmaxf(a.x, b.x);
    r.y = fmaxf(a.y, b.y);
    r.z = fmaxf(a.z, b.z);
    r.w = fmaxf(a.w, b.w);
    return r;
}

// One row (b,a) per threadIdx.y; 64 lanes in x each own one float4 of the
// 256-float feature vector. Wave32: a wave covers lanes 0..31 of one row,
// so the edge list is wave-uniform -> stage it through LDS once per row.
__global__ __launch_bounds__(256) void neural_graph_pool_kernel(
    const float* __restrict__ atoms,   // (B, A, F)
    const int*   __restrict__ edges,   // (B, A, D)
    float*       __restrict__ out)     // (B, A, F)
{
    const int tx  = threadIdx.x;                       // 0..63  float4 slot
    const int ty  = threadIdx.y;                       // 0..3   row in block
    const int row = blockIdx.x * ROWS_PER_BLOCK + ty;  // 0..B*A-1

    __shared__ int s_edges[ROWS_PER_BLOCK][ND];
    if (tx < ND) {
        s_edges[ty][tx] = edges[(size_t)row * ND + tx];
    }
    __syncthreads();

    const int b = row >> 7;        // row / NA
    const int a = row & (NA - 1);  // row % NA

    // Base of this batch's atom block, viewed as float4.
    const float4* __restrict__ batch =
        (const float4*)atoms + (size_t)b * NA * NF4;

    // include_self=True: start the max from the atom's own features.
    float4 acc = batch[(size_t)a * NF4 + tx];

    int deg = 0;
#pragma unroll
    for (int d = 0; d < ND; ++d) {
        const int e     = s_edges[ty][d];
        const bool val  = (e >= 0);
        const int  ec   = val ? e : 0;      // clamp: branch-free gather
        float4 v = batch[(size_t)ec * NF4 + tx];
        // Padding (-1) maps to the zero row in the reference -> max with 0.
        if (!val) { v.x = 0.f; v.y = 0.f; v.z = 0.f; v.w = 0.f; }
        deg += val ? 1 : 0;
        acc = f4max(acc, v);
    }

    // Atoms with zero degree are masked to zero.
    if (deg == 0) { acc.x = 0.f; acc.y = 0.f; acc.z = 0.f; acc.w = 0.f; }

    ((float4*)out)[(size_t)row * NF4 + tx] = acc;
}

extern "C" void kernel_launch(void* const* d_in, const int* in_sizes, int n_in,
                              void* d_out, int out_size, void* d_ws, size_t ws_size,
                              hipStream_t stream) {
    (void)in_sizes; (void)n_in; (void)d_ws; (void)ws_size; (void)out_size;

    const float* atoms = (const float*)d_in[0];   // (B, A, F) f32
    // d_in[1] = bonds: unused by the reference computation.
    const int*   edges = (const int*)d_in[2];     // (B, A, D) i32
    float*       out   = (float*)d_out;           // (B, A, F) f32

    dim3 block(NF4, ROWS_PER_BLOCK, 1);                  // 64 x 4 = 256 threads
    dim3 grid((NB * NA) / ROWS_PER_BLOCK, 1, 1);         // 4096 blocks

    neural_graph_pool_kernel<<<grid, block, 0, stream>>>(atoms, edges, out);
}